// LuongAttention_79791902425315
// MI455X (gfx1250) — compile-verified
//
#include <hip/hip_runtime.h>
#include <hip/hip_bf16.h>

// Problem constants (from reference setup_inputs)
#define B 32
#define S 4096
#define H 1024
#define NEG_INF (-1.0e9f)
#define NCHUNK 16
#define CHUNK_S (S / NCHUNK)   // 256 rows of keys per workgroup chunk
#define FNEG_MAX (-3.402823466e38f)

typedef __attribute__((ext_vector_type(2))) float v2f;
typedef __attribute__((ext_vector_type(8))) float v8f;

// ---------------------------------------------------------------------------
// Kernel 1: q = query @ W^T   (M=32, N=1024, K=1024) via V_WMMA_F32_16X16X4_F32
// grid.x = N/16 = 64 tiles; block = 64 threads = 2 waves; wave w owns m-tile w.
// A layout (16x4 f32): lanes 0-15 hold {A[m=lane][k], A[m=lane][k+1]},
//                      lanes 16-31 hold {A[m=lane-16][k+2], A[m=lane-16][k+3]}.
// B layout (4x16 f32) mirrors A with rows=K, cols=N.  B[k][n] = W[n*H + k].
// D layout (16x16 f32): VGPR r holds M = r + 8*(lane>=16), N = lane%16.
// ---------------------------------------------------------------------------
__global__ __launch_bounds__(64) void qproj_wmma_kernel(
    const float* __restrict__ query,   // [B, H]  (B=32 rows, the "1" dim squeezed)
    const float* __restrict__ Wq,      // [H, H]  row-major [out, in]
    float* __restrict__ qproj)         // [B, H]
{
    const int lane = threadIdx.x & 31;
    const int wave = threadIdx.x >> 5;
    const int half = lane >> 4;        // 0: K pair {0,1}, 1: K pair {2,3}
    const int lm   = lane & 15;
    const int n0   = blockIdx.x * 16;
    const int m0   = wave * 16;

    const float* aptr = query + (size_t)(m0 + lm) * H + 2 * half;
    const float* bptr = Wq    + (size_t)(n0 + lm) * H + 2 * half;

    v8f acc = {};
    for (int k = 0; k < H; k += 4) {
        v2f a, b;
        a.x = aptr[k];     a.y = aptr[k + 1];
        b.x = bptr[k];     b.y = bptr[k + 1];
        // 8 args: (neg_a, A, neg_b, B, c_mod, C, reuse_a, reuse_b)
        acc = __builtin_amdgcn_wmma_f32_16x16x4_f32(
            false, a, false, b, (short)0, acc, false, false);
    }
#pragma unroll
    for (int r = 0; r < 8; ++r) {
        qproj[(size_t)(m0 + r + 8 * half) * H + (n0 + lm)] = acc[r];
    }
}

// ---------------------------------------------------------------------------
// Kernel 2: flash-decode chunk. grid = (B, NCHUNK), block = 256 (8 wave32).
// Phase 1: each wave computes dot(q, keys[row]) for 32 rows (lane-split over H,
//          coalesced float4 loads, shfl_xor reduction). Raw masked scores are
//          kept in LDS and spilled to d_ws (needed later for `weights`).
// Then: workgroup max + exp-sum (local softmax stats).
// Phase 2: unnormalized partial context; wave w owns h-slice [128w,128w+128),
//          re-reading the same 1MB chunk (L2-resident) with coalesced loads.
// ---------------------------------------------------------------------------
__global__ __launch_bounds__(256) void attn_chunk_kernel(
    const float* __restrict__ qproj,    // [B, H]
    const float* __restrict__ keys,     // [B, S, H]
    const int*   __restrict__ src_mask, // [B, S]  (nonzero -> masked)
    float* __restrict__ scores,         // [B, S]  raw masked scores (ws)
    float* __restrict__ stats,          // [B, NCHUNK, 2] (m_local, sum_local)
    float* __restrict__ ctx_part)       // [B, NCHUNK, H] unnormalized partials
{
    const int b      = blockIdx.x;
    const int c      = blockIdx.y;
    const int s_base = c * CHUNK_S;
    const int tid    = threadIdx.x;
    const int lane   = tid & 31;
    const int wave   = tid >> 5;

    __shared__ float q_lds[H];        // 4 KB
    __shared__ float sc[CHUNK_S];     // 1 KB
    __shared__ float red[8];

    for (int i = tid; i < H; i += 256) q_lds[i] = qproj[(size_t)b * H + i];
    __syncthreads();

    const float* kb = keys + (size_t)b * S * H;

    // ---- Phase 1: scores for 256 rows (wave w -> rows w, w+8, ..., w+248)
    for (int j = 0; j < CHUNK_S / 8; ++j) {
        const int sl = wave + 8 * j;
        const int s  = s_base + sl;
        const float4* krow = (const float4*)(kb + (size_t)s * H);
        const float4* qrow = (const float4*)q_lds;
        float acc = 0.0f;
#pragma unroll
        for (int i = 0; i < H / 128; ++i) {     // 8 x float4 per lane
            float4 kv = krow[lane + 32 * i];
            float4 qv = qrow[lane + 32 * i];
            acc += kv.x * qv.x + kv.y * qv.y + kv.z * qv.z + kv.w * qv.w;
        }
#pragma unroll
        for (int off = 16; off > 0; off >>= 1)
            acc += __shfl_xor(acc, off, 32);
        if (lane == 0) {
            float sval = src_mask[(size_t)b * S + s] ? NEG_INF : acc;
            sc[sl] = sval;
            scores[(size_t)b * S + s] = sval;
        }
    }
    __syncthreads();

    // ---- local max over the 256 scores (one score per thread)
    float m = sc[tid];
#pragma unroll
    for (int off = 16; off > 0; off >>= 1)
        m = fmaxf(m, __shfl_xor(m, off, 32));
    if (lane == 0) red[wave] = m;
    __syncthreads();
    m = red[0];
#pragma unroll
    for (int w = 1; w < 8; ++w) m = fmaxf(m, red[w]);
    __syncthreads();                      // red about to be reused

    // ---- exp + local sum; overwrite sc[] with p values
    float p = __expf(sc[tid] - m);
    sc[tid] = p;
    float lsum = p;
#pragma unroll
    for (int off = 16; off > 0; off >>= 1)
        lsum += __shfl_xor(lsum, off, 32);
    if (lane == 0) red[wave] = lsum;
    __syncthreads();
    if (tid == 0) {
        float Zl = 0.0f;
#pragma unroll
        for (int w = 0; w < 8; ++w) Zl += red[w];
        stats[((size_t)b * NCHUNK + c) * 2 + 0] = m;
        stats[((size_t)b * NCHUNK + c) * 2 + 1] = Zl;
    }
    __syncthreads();                      // sc[] fully written for phase 2

    // ---- Phase 2: partial context. wave owns h in [wave*128, wave*128+128)
    const float* kcol = kb + (size_t)s_base * H + wave * 128 + lane * 4;
    float4 acc4 = {0.0f, 0.0f, 0.0f, 0.0f};
#pragma unroll 4
    for (int sl = 0; sl < CHUNK_S; ++sl) {
        float  pv = sc[sl];
        float4 kv = *(const float4*)(kcol + (size_t)sl * H);
        acc4.x += pv * kv.x;
        acc4.y += pv * kv.y;
        acc4.z += pv * kv.z;
        acc4.w += pv * kv.w;
    }
    float* cp = ctx_part + ((size_t)b * NCHUNK + c) * H + wave * 128 + lane * 4;
    *(float4*)cp = acc4;
}

// ---------------------------------------------------------------------------
// Kernel 3: combine chunks. grid = B, block = 256.
// Global max/Z from 16 chunk stats (computed redundantly per thread - trivial),
// then write context [B,H] and weights [B,S] (all of d_out).
// ---------------------------------------------------------------------------
__global__ __launch_bounds__(256) void combine_kernel(
    const float* __restrict__ scores,    // [B, S]
    const float* __restrict__ stats,     // [B, NCHUNK, 2]
    const float* __restrict__ ctx_part,  // [B, NCHUNK, H]
    float* __restrict__ out_ctx,         // [B, H]
    float* __restrict__ out_w)           // [B, S]
{
    const int b   = blockIdx.x;
    const int tid = threadIdx.x;

    float mg = FNEG_MAX;
#pragma unroll
    for (int c = 0; c < NCHUNK; ++c)
        mg = fmaxf(mg, stats[((size_t)b * NCHUNK + c) * 2 + 0]);

    float Z = 0.0f;
    float scale[NCHUNK];
#pragma unroll
    for (int c = 0; c < NCHUNK; ++c) {
        float e = __expf(stats[((size_t)b * NCHUNK + c) * 2 + 0] - mg);
        scale[c] = e;
        Z += stats[((size_t)b * NCHUNK + c) * 2 + 1] * e;
    }
    const float invZ = 1.0f / Z;

    for (int h = tid; h < H; h += 256) {
        float acc = 0.0f;
#pragma unroll
        for (int c = 0; c < NCHUNK; ++c)
            acc += ctx_part[((size_t)b * NCHUNK + c) * H + h] * scale[c];
        out_ctx[(size_t)b * H + h] = acc * invZ;
    }
    for (int s = tid; s < S; s += 256) {
        out_w[(size_t)b * S + s] = __expf(scores[(size_t)b * S + s] - mg) * invZ;
    }
}

// ---------------------------------------------------------------------------
extern "C" void kernel_launch(void* const* d_in, const int* in_sizes, int n_in,
                              void* d_out, int out_size, void* d_ws, size_t ws_size,
                              hipStream_t stream) {
    const float* query = (const float*)d_in[0];   // [B,1,H] fp32
    const float* keys  = (const float*)d_in[1];   // [B,S,H] fp32
    const int*   mask  = (const int*)  d_in[2];   // [B,S]   bool -> int
    const float* Wq    = (const float*)d_in[3];   // [H,H]   fp32

    float* out_ctx = (float*)d_out;               // first B*H floats
    float* out_w   = out_ctx + (size_t)B * H;     // next  B*S floats

    // Workspace layout (floats): qproj | scores | stats | ctx_part  (~2.7 MB)
    float* ws       = (float*)d_ws;
    float* qproj    = ws;                                  // B*H
    float* scores   = qproj    + (size_t)B * H;            // B*S
    float* stats    = scores   + (size_t)B * S;            // B*NCHUNK*2
    float* ctx_part = stats    + (size_t)B * NCHUNK * 2;   // B*NCHUNK*H

    // 1) projection GEMM on the matrix engine
    qproj_wmma_kernel<<<dim3(H / 16), 64, 0, stream>>>(query, Wq, qproj);
    // 2) single streaming pass over keys (HBM-bound: ~512 MB @ 23.3 TB/s)
    attn_chunk_kernel<<<dim3(B, NCHUNK), 256, 0, stream>>>(
        qproj, keys, mask, scores, stats, ctx_part);
    // 3) merge chunk stats, emit context + weights
    combine_kernel<<<dim3(B), 256, 0, stream>>>(
        scores, stats, ctx_part, out_ctx, out_w);
}